// Net_79534204388010
// MI455X (gfx1250) — compile-verified
//
#include <hip/hip_runtime.h>
#include <hip/hip_bf16.h>

// ---------------------------------------------------------------------------
// 2-layer GCN link prediction for MI455X (gfx1250, wave32).
// Dense parts (x@W1, h@W2) use V_WMMA_F32_16X16X4_F32 (full fp32 precision).
// Sparse parts (degree, scatter-add, link dot) are memory/L2-atomic bound:
// float4-vectorized gathers, atomics are the irreducible cost (L2-resident).
// ---------------------------------------------------------------------------

typedef float v2f __attribute__((ext_vector_type(2)));
typedef float v8f __attribute__((ext_vector_type(8)));

// ---------------- degree / normalization ----------------

__global__ void deg_init_kernel(float* __restrict__ deg, int n) {
    int i = blockIdx.x * blockDim.x + threadIdx.x;
    if (i < n) deg[i] = 1.0f;               // self-loop contribution
}

__global__ void deg_count_kernel(const long long* __restrict__ dst, int E,
                                 float* __restrict__ deg) {
    int e = blockIdx.x * blockDim.x + threadIdx.x;
    if (e < E) atomicAdd(&deg[(int)dst[e]], 1.0f);
}

__global__ void dinv_kernel(float* __restrict__ deg, int n) {
    int i = blockIdx.x * blockDim.x + threadIdx.x;
    if (i < n) deg[i] = rsqrtf(deg[i]);     // deg >= 1 always (self-loop)
}

// ---------------- fp32 WMMA GEMM: H[nrows x NCOLS] = X[nrows x 64] @ W[64 x NCOLS]
// One wave computes a 16x16 tile of H via 16 chained v_wmma_f32_16x16x4_f32.
// A (16x4 f32): lanes 0-15 -> M=0..15, v0=K0 v1=K1 ; lanes 16-31 -> v0=K2 v1=K3.
// B (4x16 f32): mirrored layout (lane group selects K pair, lane%16 = N).
// C/D (16x16 f32): 8 VGPRs, lane<16 -> M=r, lane>=16 -> M=8+r, N = lane%16.

template <int NCOLS>
__global__ __launch_bounds__(256)
void gemm_wmma_kernel(const float* __restrict__ X, const float* __restrict__ W,
                      float* __restrict__ H, int nrows) {
    constexpr int K = 64;
    constexpr int nColTiles = NCOLS / 16;
    const int wavesPerBlock = blockDim.x >> 5;
    const int waveId = blockIdx.x * wavesPerBlock + ((int)threadIdx.x >> 5);
    const int lane = (int)threadIdx.x & 31;

    const int nRowTiles = nrows >> 4;           // nrows % 16 == 0 (50000/16 = 3125)
    if (waveId >= nRowTiles * nColTiles) return; // uniform per wave: EXEC stays full

    const int rowTile = waveId / nColTiles;
    const int colTile = waveId % nColTiles;
    const int row0 = rowTile << 4;
    const int col0 = colTile << 4;

    const int m    = lane & 15;                 // M (for A) / N (for B, C)
    const int hi   = lane >> 4;                 // lane group
    const int koff = hi * 2;                    // K offset within the 4-wide slab

    const float* __restrict__ xrow = X + (size_t)(row0 + m) * K;

    v8f acc = {};
#pragma unroll
    for (int k = 0; k < K; k += 4) {
        v2f a;
        a.x = xrow[k + koff + 0];
        a.y = xrow[k + koff + 1];
        v2f b;
        b.x = W[(size_t)(k + koff + 0) * NCOLS + col0 + m];
        b.y = W[(size_t)(k + koff + 1) * NCOLS + col0 + m];
        acc = __builtin_amdgcn_wmma_f32_16x16x4_f32(
            /*neg_a=*/false, a, /*neg_b=*/false, b,
            /*c_mod=*/(short)0, acc, /*reuse_a=*/false, /*reuse_b=*/false);
    }

    const int rbase = row0 + hi * 8;
#pragma unroll
    for (int r = 0; r < 8; ++r)
        H[(size_t)(rbase + r) * NCOLS + col0 + m] = acc[r];
}

// ---------------- GCN aggregation (float4-vectorized) ----------------

// out[i][4f..4f+3] = dinv[i]^2 * h[i][..] + bias[..]   (self-loop message + bias)
template <int C>   // C = channels, multiple of 4
__global__ void init_out_kernel(const float4* __restrict__ h,
                                const float* __restrict__ dinv,
                                const float4* __restrict__ bias,
                                float4* __restrict__ out, int n) {
    constexpr int CV = C / 4;
    int idx = blockIdx.x * blockDim.x + threadIdx.x;   // over n*CV float4s
    if (idx >= n * CV) return;
    int node = idx / CV;
    int f    = idx - node * CV;
    float w  = dinv[node];
    w *= w;
    float4 hv = h[idx], bv = bias[f];
    float4 o;
    o.x = w * hv.x + bv.x;
    o.y = w * hv.y + bv.y;
    o.z = w * hv.z + bv.z;
    o.w = w * hv.w + bv.w;
    out[idx] = o;
}

// out[dst][4f..4f+3] += dinv[src]*dinv[dst] * h[src][..]
// One thread per (edge, float4-chunk): index loads amortized 4x, gather of
// h[src] is a coalesced 128-bit load; atomics remain the irreducible cost.
template <int C>
__global__ void scatter_kernel(const long long* __restrict__ src,
                               const long long* __restrict__ dst, int E,
                               const float4* __restrict__ h,
                               const float* __restrict__ dinv,
                               float* __restrict__ out) {
    constexpr int CV = C / 4;
    long long idx = (long long)blockIdx.x * blockDim.x + threadIdx.x;
    if (idx >= (long long)E * CV) return;
    int e = (int)(idx / CV);
    int f = (int)(idx - (long long)e * CV);
    int s = (int)src[e];
    int d = (int)dst[e];
    float w = dinv[s] * dinv[d];
    float4 m = h[(size_t)s * CV + f];
    float* o = out + (size_t)d * C + 4 * f;
    atomicAdd(o + 0, w * m.x);
    atomicAdd(o + 1, w * m.y);
    atomicAdd(o + 2, w * m.z);
    atomicAdd(o + 3, w * m.w);
}

__global__ void relu_kernel(float4* __restrict__ x, int n4) {
    int i = blockIdx.x * blockDim.x + threadIdx.x;
    if (i >= n4) return;
    float4 v = x[i];
    v.x = fmaxf(v.x, 0.0f);
    v.y = fmaxf(v.y, 0.0f);
    v.z = fmaxf(v.z, 0.0f);
    v.w = fmaxf(v.w, 0.0f);
    x[i] = v;
}

// ---------------- link scoring: pred[l] = dot(h2[a[l]], h2[b[l]]) over 32 ch ----

__global__ void score_kernel(const long long* __restrict__ a,
                             const long long* __restrict__ b, int L,
                             const float* __restrict__ h2,
                             float* __restrict__ pred) {
    int l = blockIdx.x * blockDim.x + threadIdx.x;
    if (l >= L) return;
    const float4* pa = (const float4*)(h2 + (size_t)a[l] * 32);
    const float4* pb = (const float4*)(h2 + (size_t)b[l] * 32);
    float s = 0.0f;
#pragma unroll
    for (int i = 0; i < 8; ++i) {
        float4 u = pa[i], v = pb[i];
        s += u.x * v.x + u.y * v.y + u.z * v.z + u.w * v.w;
    }
    pred[l] = s;
}

// ---------------------------------------------------------------------------

static inline int cdiv(long long a, int b) { return (int)((a + b - 1) / b); }

extern "C" void kernel_launch(void* const* d_in, const int* in_sizes, int n_in,
                              void* d_out, int out_size, void* d_ws, size_t ws_size,
                              hipStream_t stream) {
    const float*     x   = (const float*)d_in[0];       // [N,64]
    const long long* ei  = (const long long*)d_in[1];   // [2,E] int64
    const long long* eli = (const long long*)d_in[2];   // [2,L] int64
    const float*     W1  = (const float*)d_in[3];       // [64,64]
    const float*     b1  = (const float*)d_in[4];       // [64]
    const float*     W2  = (const float*)d_in[5];       // [64,32]
    const float*     b2  = (const float*)d_in[6];       // [32]
    float*           pred = (float*)d_out;              // [L]

    const int N = in_sizes[0] / 64;                     // 50000
    const int E = in_sizes[1] / 2;                      // 800000
    const int L = in_sizes[2] / 2;                      // 200000

    const long long* esrc = ei;
    const long long* edst = ei + E;
    const long long* la   = eli;
    const long long* lb   = eli + L;

    // workspace layout (floats, 16B-aligned blocks)
    float* ws   = (float*)d_ws;
    size_t off  = 0;
    float* dinv = ws + off; off += (size_t)((N + 63) & ~63);   // 50048
    float* h1   = ws + off; off += (size_t)N * 64;             // 3.2M
    float* out1 = ws + off; off += (size_t)N * 64;             // 3.2M
    float* h2   = ws + off; off += (size_t)N * 32;             // 1.6M
    float* out2 = ws + off; off += (size_t)N * 32;             // 1.6M
    (void)ws_size;

    const int B = 256;

    // 1) degrees -> dinv
    deg_init_kernel<<<cdiv(N, B), B, 0, stream>>>(dinv, N);
    deg_count_kernel<<<cdiv(E, B), B, 0, stream>>>(edst, E, dinv);
    dinv_kernel<<<cdiv(N, B), B, 0, stream>>>(dinv, N);

    // 2) layer 1: h1 = x @ W1   (WMMA fp32), then aggregate + ReLU
    {
        int waves = (N / 16) * (64 / 16);               // 12500
        gemm_wmma_kernel<64><<<cdiv(waves, B / 32), B, 0, stream>>>(x, W1, h1, N);
    }
    init_out_kernel<64><<<cdiv((long long)N * 16, B), B, 0, stream>>>(
        (const float4*)h1, dinv, (const float4*)b1, (float4*)out1, N);
    scatter_kernel<64><<<cdiv((long long)E * 16, B), B, 0, stream>>>(
        esrc, edst, E, (const float4*)h1, dinv, out1);
    relu_kernel<<<cdiv((long long)N * 16, B), B, 0, stream>>>((float4*)out1, N * 16);

    // 3) layer 2: h2 = out1 @ W2 (WMMA fp32), then aggregate (no ReLU)
    {
        int waves = (N / 16) * (32 / 16);               // 6250
        gemm_wmma_kernel<32><<<cdiv(waves, B / 32), B, 0, stream>>>(out1, W2, h2, N);
    }
    init_out_kernel<32><<<cdiv((long long)N * 8, B), B, 0, stream>>>(
        (const float4*)h2, dinv, (const float4*)b2, (float4*)out2, N);
    scatter_kernel<32><<<cdiv((long long)E * 8, B), B, 0, stream>>>(
        esrc, edst, E, (const float4*)h2, dinv, out2);

    // 4) link scoring
    score_kernel<<<cdiv(L, B), B, 0, stream>>>(la, lb, L, out2, pred);
}